// FFM_66383014527546
// MI455X (gfx1250) — compile-verified
//
#include <hip/hip_runtime.h>
#include <hip/hip_bf16.h>
#include <stdint.h>

// FFM pairwise interaction:
//   out[b, p, e] = x[b,i(p),e] * x[b,j(p),e] * fe[i,j,e] * fe[j,i,e]
// B=4096, F=40, E=64, P=780. Pure elementwise -> store-bandwidth bound
// (~818 MB output, floor ~37us @ 23.3 TB/s). No contraction -> WMMA N/A.
// CDNA5 paths used: TDM tensor_load_to_lds (async DMA of x-row into LDS),
// s_wait_tensorcnt, and non-temporal b128 output stores.

typedef float v4f  __attribute__((ext_vector_type(4)));
typedef unsigned u32x4 __attribute__((ext_vector_type(4)));
typedef int  i32x8 __attribute__((ext_vector_type(8)));
typedef int  i32x4 __attribute__((ext_vector_type(4)));

#define NUM_FIELDS 40
#define EMBED      64
#define NPAIRS     780           // 40*39/2
#define THREADS    256
#define XROW_ELTS  (NUM_FIELDS * EMBED)        // 2560 floats = 10240 B
#define XROW_V4    (XROW_ELTS / 4)             // 640 float4

__global__ __launch_bounds__(THREADS)
void ffm_pair_kernel(const float* __restrict__ x,   // [B, 40, 64]
                     const float* __restrict__ fe,  // [40, 40, 64]
                     float* __restrict__ out)       // [B, 780*64]
{
    __shared__ v4f      lds_x[XROW_V4];    // 10 KB: x[b,:,:]
    __shared__ unsigned pair_tab[NPAIRS];  // (i<<8)|j per pair

    const int    b    = blockIdx.x;
    const int    tid  = threadIdx.x;
    const size_t xrow = (size_t)b * XROW_ELTS;

    // ---- Stage x[b,:,:] into LDS ----
#if defined(__gfx1250__) && __has_builtin(__builtin_amdgcn_tensor_load_to_lds)
    if (tid == 0) {
        // Tensor DMA descriptor (D#), 2D tensor: 2560 x 1 elements of 4 bytes.
        const uint64_t ga = (uint64_t)(uintptr_t)(x + xrow);
        const uint32_t lo = (uint32_t)(uintptr_t)(void*)&lds_x[0]; // LDS byte offset

        u32x4 g0;
        g0.x = 1u;                                   // count=1 (valid), user mode
        g0.y = lo;                                   // lds_addr
        g0.z = (uint32_t)(ga & 0xFFFFFFFFu);         // global_addr[31:0]
        g0.w = (uint32_t)((ga >> 32) & 0x01FFFFFFu)  // global_addr[56:32]
             | (2u << 30);                           // type = 2 ("image")

        i32x8 g1;
        g1[0] = (int)(2u << 16);                       // data_size = 2 -> 4 bytes
        g1[1] = (int)((XROW_ELTS & 0xFFFFu) << 16);    // tensor_dim0 lo16 = 2560
        g1[2] = (int)((XROW_ELTS >> 16) | (1u << 16)); // td0 hi16 | tensor_dim1 lo16 = 1
        g1[3] = (int)((unsigned)XROW_ELTS << 16);      // td1 hi16=0 | tile_dim0 = 2560
        g1[4] = 1;                                     // tile_dim1 = 1, tile_dim2 = 0
        g1[5] = XROW_ELTS;                             // tensor_dim0_stride lo32
        g1[6] = 0;                                     // stride hi | dim1_stride lo
        g1[7] = 0;

        i32x4 z4 = {0, 0, 0, 0};                       // groups 2/3 unused (2D)
        i32x8 z8 = {0, 0, 0, 0, 0, 0, 0, 0};           // extra group (6-arg form)
        __builtin_amdgcn_tensor_load_to_lds(g0, g1, z4, z4, z8, 0);
    }
    __builtin_amdgcn_s_wait_tensorcnt(0);
#else
    {
        const v4f* x4 = (const v4f*)(x + xrow);
        for (int t = tid; t < XROW_V4; t += THREADS) lds_x[t] = x4[t];
    }
#endif

    // ---- Pair index table: triu(F, k=1) in row-major (i<j) order ----
    for (int p = tid; p < NPAIRS; p += THREADS) {
        int i = 0, rem = p;
        while (rem >= (NUM_FIELDS - 1 - i)) { rem -= (NUM_FIELDS - 1 - i); ++i; }
        pair_tab[p] = (unsigned)((i << 8) | (i + 1 + rem));
    }
    __syncthreads();

    // ---- Main streaming loop: 16 lanes per pair, 16 pairs per iteration ----
    const v4f*   fe4     = (const v4f*)fe;
    const int    sub     = tid >> 4;               // pair slot 0..15
    const int    lane    = tid & 15;               // float4 slot within E=64
    const size_t outbase = (size_t)b * (NPAIRS * (EMBED / 4));
    v4f*         out4    = (v4f*)out;

    for (int p0 = 0; p0 < NPAIRS; p0 += 16) {
        const int p = p0 + sub;
        if (p < NPAIRS) {
            const unsigned ij = pair_tab[p];
            const int i = (int)(ij >> 8);
            const int j = (int)(ij & 0xFFu);

            const v4f xi  = lds_x[(i << 4) + lane];
            const v4f xj  = lds_x[(j << 4) + lane];
            const v4f fij = fe4[((size_t)(i * NUM_FIELDS + j) << 4) + lane];
            const v4f fji = fe4[((size_t)(j * NUM_FIELDS + i) << 4) + lane];

            const v4f r = (xi * xj) * (fij * fji);
            // 818 MB write-once stream: non-temporal b128, fully coalesced.
            __builtin_nontemporal_store(r, &out4[outbase + ((size_t)p << 4) + lane]);
        }
    }
}

extern "C" void kernel_launch(void* const* d_in, const int* in_sizes, int n_in,
                              void* d_out, int out_size, void* d_ws, size_t ws_size,
                              hipStream_t stream) {
    const float* x  = (const float*)d_in[0];   // [B, 40, 64] fp32
    const float* fe = (const float*)d_in[1];   // [40, 40, 64] fp32
    float*       o  = (float*)d_out;           // [B, 780*64] fp32

    const int B = in_sizes[0] / XROW_ELTS;     // 4096
    ffm_pair_kernel<<<dim3(B), dim3(THREADS), 0, stream>>>(x, fe, o);
}